// ResidualBlock_57277683859710
// MI455X (gfx1250) — compile-verified
//
#include <hip/hip_runtime.h>
#include <hip/hip_bf16.h>

typedef __attribute__((ext_vector_type(2))) float v2f;
typedef __attribute__((ext_vector_type(4))) float v4f;
typedef __attribute__((ext_vector_type(8))) float v8f;

#ifndef __has_builtin
#define __has_builtin(x) 0
#endif

#if defined(__HIP_DEVICE_COMPILE__) && __has_builtin(__builtin_amdgcn_global_load_async_to_lds_b32)
#define HAVE_ASYNC_LDS 1
#else
#define HAVE_ASYNC_LDS 0
#endif

typedef __attribute__((address_space(1))) int gas_int;   // global
typedef __attribute__((address_space(3))) int las_int;   // LDS

namespace {
constexpr int DIL   = 512;
constexpr int CRES  = 128;
constexpr int CSKIP = 256;
constexpr int TLEN  = 16384;
constexpr int LLEN  = TLEN - DIL;           // 15872
constexpr int NT    = 64;                   // time tile per workgroup
constexpr int LDST  = CRES + 4;             // padded LDS row stride (floats) = 132
constexpr int SKIP_LEN   = 4096;
constexpr int SKIP_START = LLEN - SKIP_LEN; // 11776 (multiple of NT)
constexpr int NJ = NT / 16;                 // 4 n-tiles per wave
constexpr size_t WS_NEED = (size_t)2 * CRES * CRES * sizeof(float);
constexpr int LDS_BYTES = 3 * NT * LDST * (int)sizeof(float);   // ~99 KB dynamic LDS
}

__device__ __forceinline__ float gated_act(float p) {
    // tanh(p) * sigmoid(p) via e = exp(-p): tanh = (1-e^2)/(1+e^2), sig = 1/(1+e)
    p = fminf(fmaxf(p, -30.0f), 30.0f);
    float e  = __expf(-p);
    float e2 = e * e;
    float th = (1.0f - e2) * __builtin_amdgcn_rcpf(1.0f + e2);
    float sg = __builtin_amdgcn_rcpf(1.0f + e);
    return th * sg;
}

__device__ __forceinline__ void async_wait_all() {
#if HAVE_ASYNC_LDS
#if __has_builtin(__builtin_amdgcn_s_wait_asynccnt)
    __builtin_amdgcn_s_wait_asynccnt(0);
#else
    asm volatile("s_wait_asynccnt 0x0" ::: "memory");
#endif
#endif
}

// Deinterleave w_dil (o,i,tap) -> W0[o][i], W1[o][i] row-major in workspace.
__global__ void deinterleave_wdil(const float* __restrict__ w_dil,
                                  float* __restrict__ w0,
                                  float* __restrict__ w1) {
    int i = blockIdx.x * 256 + threadIdx.x;       // 0 .. 16383
    float a = w_dil[2 * i + 0];
    float b = w_dil[2 * i + 1];
    w0[i] = a;
    w1[i] = b;
}

template <bool DEINT>
__global__ __launch_bounds__(256)
void wavenet_residual_block(const float* __restrict__ x,
                            const float* __restrict__ w_dil,   // interleaved (fallback)
                            const float* __restrict__ w0,      // deinterleaved tap 0
                            const float* __restrict__ w1,      // deinterleaved tap 1
                            const float* __restrict__ w_res,
                            const float* __restrict__ b_res,
                            const float* __restrict__ w_skip,
                            const float* __restrict__ b_skip,
                            float* __restrict__ out,
                            float* __restrict__ skip)
{
    extern __shared__ float smem[];
    float* lds_x0  = smem;                  // x[b, :, t0 .. t0+NT)     as [t][c]
    float* lds_x1  = smem +     NT * LDST;  // x[b, :, t0+512 .. +NT)   as [t][c]
    float* lds_act = smem + 2 * NT * LDST;  // gated activation         as [t][c]

    const int tid  = threadIdx.x;
    const int wave = tid >> 5;
    const int lane = tid & 31;
    const int lh   = lane >> 4;   // half-wave select
    const int l16  = lane & 15;

    const int t0 = blockIdx.x * NT;
    const int b  = blockIdx.y;

    const float* xb = x + (size_t)b * CRES * TLEN;

    // ---- stage 0: global -> LDS (transposed to [t][c]) ----
#if HAVE_ASYNC_LDS
    #pragma unroll
    for (int i = 0; i < (CRES * NT) / 256; ++i) {
        int idx = tid + i * 256;
        int c = idx >> 6;               // idx / NT
        int t = idx & (NT - 1);
        __builtin_amdgcn_global_load_async_to_lds_b32(
            (gas_int*)(xb + (size_t)c * TLEN + t0 + t),
            (las_int*)&lds_x0[t * LDST + c], 0, 0);
        __builtin_amdgcn_global_load_async_to_lds_b32(
            (gas_int*)(xb + (size_t)c * TLEN + t0 + DIL + t),
            (las_int*)&lds_x1[t * LDST + c], 0, 0);
    }
    async_wait_all();
#else
    #pragma unroll
    for (int i = 0; i < (CRES * NT) / 256; ++i) {
        int idx = tid + i * 256;
        int c = idx >> 6;
        int t = idx & (NT - 1);
        lds_x0[t * LDST + c] = xb[(size_t)c * TLEN + t0 + t];
        lds_x1[t * LDST + c] = xb[(size_t)c * TLEN + t0 + DIL + t];
    }
#endif
    __syncthreads();

    const int mbase = wave * 16;   // wave owns output rows [mbase, mbase+16)

    // ---- stage 1: pre = W0 @ x0 + W1 @ x1 ----
    v8f acc[NJ] = {};
    #pragma unroll 4
    for (int ks = 0; ks < CRES / 4; ++ks) {
        const int kk = 4 * ks + 2 * lh;
        v2f a0, a1;
        if constexpr (DEINT) {
            a0 = *(const v2f*)(w0 + (size_t)(mbase + l16) * CRES + kk);
            a1 = *(const v2f*)(w1 + (size_t)(mbase + l16) * CRES + kk);
        } else {
            v4f wa = *(const v4f*)(w_dil + ((size_t)(mbase + l16) * CRES + kk) * 2);
            a0 = v2f{ wa.x, wa.z };
            a1 = v2f{ wa.y, wa.w };
        }
        #pragma unroll
        for (int j = 0; j < NJ; ++j) {
            v2f b0 = *(const v2f*)(lds_x0 + (j * 16 + l16) * LDST + kk);
            v2f b1 = *(const v2f*)(lds_x1 + (j * 16 + l16) * LDST + kk);
            acc[j] = __builtin_amdgcn_wmma_f32_16x16x4_f32(false, a0, false, b0,
                                                           (short)0, acc[j], false, false);
            acc[j] = __builtin_amdgcn_wmma_f32_16x16x4_f32(false, a1, false, b1,
                                                           (short)0, acc[j], false, false);
        }
    }

    // ---- gate: act = tanh(pre)*sigmoid(pre), stored B-ready [t][c] ----
    #pragma unroll
    for (int j = 0; j < NJ; ++j) {
        const int n = j * 16 + l16;
        #pragma unroll
        for (int r = 0; r < 8; ++r) {
            lds_act[n * LDST + mbase + 8 * lh + r] = gated_act(acc[j][r]);
        }
    }
    __syncthreads();

    // ---- stage 2a: out = w_res @ act + b_res + x1 (residual) ----
    v8f oacc[NJ] = {};
    #pragma unroll 4
    for (int ks = 0; ks < CRES / 4; ++ks) {
        const int kk = 4 * ks + 2 * lh;
        v2f aw = *(const v2f*)(w_res + (size_t)(mbase + l16) * CRES + kk);
        #pragma unroll
        for (int j = 0; j < NJ; ++j) {
            v2f bb = *(const v2f*)(lds_act + (j * 16 + l16) * LDST + kk);
            oacc[j] = __builtin_amdgcn_wmma_f32_16x16x4_f32(false, aw, false, bb,
                                                            (short)0, oacc[j], false, false);
        }
    }
    {
        const float* bp = b_res + mbase + 8 * lh;   // 8 consecutive rows per lane
        v4f bia0 = *(const v4f*)(bp);
        v4f bia1 = *(const v4f*)(bp + 4);
        #pragma unroll
        for (int j = 0; j < NJ; ++j) {
            const int n = j * 16 + l16;
            const float* rp = lds_x1 + n * LDST + mbase + 8 * lh;
            v4f r0 = *(const v4f*)(rp);
            v4f r1 = *(const v4f*)(rp + 4);
            #pragma unroll
            for (int r = 0; r < 8; ++r) {
                float bias = (r < 4) ? bia0[r] : bia1[r - 4];
                float res  = (r < 4) ? r0[r]   : r1[r - 4];
                int m = mbase + 8 * lh + r;
                out[(size_t)(b * CRES + m) * LLEN + t0 + n] = oacc[j][r] + bias + res;
            }
        }
    }

    // ---- stage 2b: skip = w_skip @ act + b_skip, last 4096 cols only ----
    if (t0 >= SKIP_START) {                 // workgroup-uniform branch
        v8f sacc[2][NJ] = {};
        #pragma unroll 4
        for (int ks = 0; ks < CRES / 4; ++ks) {
            const int kk = 4 * ks + 2 * lh;
            v2f awA = *(const v2f*)(w_skip + (size_t)((wave)     * 16 + l16) * CRES + kk);
            v2f awB = *(const v2f*)(w_skip + (size_t)((wave + 8) * 16 + l16) * CRES + kk);
            #pragma unroll
            for (int j = 0; j < NJ; ++j) {
                v2f bb = *(const v2f*)(lds_act + (j * 16 + l16) * LDST + kk);
                sacc[0][j] = __builtin_amdgcn_wmma_f32_16x16x4_f32(false, awA, false, bb,
                                                                   (short)0, sacc[0][j], false, false);
                sacc[1][j] = __builtin_amdgcn_wmma_f32_16x16x4_f32(false, awB, false, bb,
                                                                   (short)0, sacc[1][j], false, false);
            }
        }
        #pragma unroll
        for (int h = 0; h < 2; ++h) {
            const int mb = (wave + 8 * h) * 16;
            const float* bp = b_skip + mb + 8 * lh;
            v4f bia0 = *(const v4f*)(bp);
            v4f bia1 = *(const v4f*)(bp + 4);
            #pragma unroll
            for (int j = 0; j < NJ; ++j) {
                const int n = j * 16 + l16;
                #pragma unroll
                for (int r = 0; r < 8; ++r) {
                    float bias = (r < 4) ? bia0[r] : bia1[r - 4];
                    int m = mb + 8 * lh + r;
                    skip[(size_t)(b * CSKIP + m) * SKIP_LEN + (t0 - SKIP_START) + n] =
                        sacc[h][j][r] + bias;
                }
            }
        }
    }
}

extern "C" void kernel_launch(void* const* d_in, const int* in_sizes, int n_in,
                              void* d_out, int out_size, void* d_ws, size_t ws_size,
                              hipStream_t stream) {
    (void)in_sizes; (void)n_in; (void)out_size;
    const float* x      = (const float*)d_in[0];
    const float* w_dil  = (const float*)d_in[1];
    const float* w_res  = (const float*)d_in[2];
    const float* b_res  = (const float*)d_in[3];
    const float* w_skip = (const float*)d_in[4];
    const float* b_skip = (const float*)d_in[5];
    // d_in[6] = skip_size scalar (fixed 4096 here)

    float* out  = (float*)d_out;
    float* skip = out + (size_t)8 * CRES * LLEN;   // outputs concatenated in return order

    dim3 grid(LLEN / NT, 8);   // 248 time tiles x 8 batches

    if (ws_size >= WS_NEED) {
        float* w0 = (float*)d_ws;
        float* w1 = w0 + CRES * CRES;
        deinterleave_wdil<<<dim3((CRES * CRES) / 256), dim3(256), 0, stream>>>(w_dil, w0, w1);
        wavenet_residual_block<true><<<grid, dim3(256), LDS_BYTES, stream>>>(
            x, w_dil, w0, w1, w_res, b_res, w_skip, b_skip, out, skip);
    } else {
        wavenet_residual_block<false><<<grid, dim3(256), LDS_BYTES, stream>>>(
            x, w_dil, w_dil, w_dil, w_res, b_res, w_skip, b_skip, out, skip);
    }
}